// BBoxAwareLoss_21784074126165
// MI455X (gfx1250) — compile-verified
//
#include <hip/hip_runtime.h>
#include <cstdint>

// Problem constants (reference: shape (32,1,1024,1024), 5x5 structuring element)
#define IMG_W   1024
#define IMG_H   1024
#define IMG_B   32
#define TW      128      // tile width (output)
#define TH      64       // tile height (output)
#define SW      144      // staged width in floats: covers x0-8 .. x0+135 (halo 2 + B128 align)
#define SR      68       // staged rows: y0-2 .. y0+65
#define NBLK    4096     // 8 * 16 * 32 blocks
#define N_TOTAL 33554432.0f

typedef __attribute__((ext_vector_type(2))) float v2f;
typedef __attribute__((ext_vector_type(8))) float v8f;

// ---------------------------------------------------------------------------
// Kernel 1: fused tile kernel.
//  - async-stage target tile (+halo) into LDS (ASYNCcnt path)
//  - bit-pack binarized mask via wave32 ballot (dilate=OR, erode=AND domain)
//  - separable 5x5 window with bitwise shifts (row pass) + word ORs/ANDs (col)
//  - stream pred with float4 loads, fuse BCE loss + edge weighting
//  - per-block partial sums (plain & weighted) + has-one/has-zero flags
// ---------------------------------------------------------------------------
__global__ __launch_bounds__(256) void edge_bce_tile_kernel(
    const float* __restrict__ pred, const float* __restrict__ target,
    float* __restrict__ wsP, float* __restrict__ wsW, unsigned* __restrict__ wsF)
{
    __shared__ float    stage[SR * SW];     // raw target floats (tile + halo)
    __shared__ unsigned bOr[SR][6];         // packed mask, OOB -> 0 (dilate-safe)
    __shared__ unsigned bAnd[SR][6];        // packed mask, OOB -> 1 (erode-safe)
    __shared__ unsigned rOr[SR][4];         // 5-wide row OR, core words
    __shared__ unsigned rAnd[SR][4];        // 5-wide row AND, core words
    __shared__ float    redP[8], redW[8];
    __shared__ unsigned redF[8];

    const int tid  = threadIdx.x;
    const int wave = tid >> 5;
    const int ln   = tid & 31;
    const int x0   = blockIdx.x * TW;
    const int y0   = blockIdx.y * TH;
    const int z    = blockIdx.z;

    const float* timg = target + ((size_t)z << 20);
    const float* pimg = pred   + ((size_t)z << 20);

    // ---- Phase 0: async bulk-fill target tile into LDS (B128 per lane) ----
    const unsigned ldsStage = (unsigned)(uintptr_t)(void*)stage; // AS(3) offset in low 32 bits
    for (int s = tid; s < SR * (SW / 4); s += 256) {             // 68*36 = 2448 x 16B slots
        const int r  = s / (SW / 4);
        const int c4 = s - r * (SW / 4);
        const int gy = y0 - 2 + r;
        const int gx = x0 - 8 + c4 * 4;
        if (gy >= 0 && gy < IMG_H && gx >= 0 && gx < IMG_W) {
            const float* gp = timg + (size_t)gy * IMG_W + gx;
            const unsigned la = ldsStage + (unsigned)(r * SW + c4 * 4) * 4u;
            asm volatile("global_load_async_to_lds_b128 %0, %1, off"
                         :: "v"(la), "v"(gp) : "memory");
        }
    }
    asm volatile("s_wait_asynccnt 0" ::: "memory");  // per-wave completion
    __syncthreads();                                  // cross-wave visibility

    // ---- Phase 1: binarize + bit-pack via wave32 ballot -------------------
    // word index wa in 0..5 maps to x-word w = wa-1 (bit i <-> gx = x0 + 32w + i)
    for (int it = wave; it < SR * 6; it += 8) {       // 408 words, 51/wave (uniform)
        const int r  = it / 6;
        const int wa = it - r * 6;
        const int gy  = y0 - 2 + r;
        const int gx  = x0 + (wa - 1) * 32 + ln;
        const int col = (wa - 1) * 32 + ln + 8;       // gx - (x0 - 8)
        const bool valid = (gy >= 0) && (gy < IMG_H) && (gx >= 0) && (gx < IMG_W)
                           && (col >= 0) && (col < SW);
        const bool bv = valid && (stage[r * SW + col] > 0.5f);
        const unsigned mOr  = (unsigned)__ballot(bv);
        const unsigned mAnd = (unsigned)__ballot(bv || !valid);
        if (ln == 0) { bOr[r][wa] = mOr; bAnd[r][wa] = mAnd; }
    }
    __syncthreads();

    // ---- Phase 2: 5-wide row window (bitwise, cross-word via neighbors) ---
    for (int i = tid; i < SR * 4; i += 256) {         // 272 core words
        const int r = i >> 2, w = i & 3, wa = w + 1;
        {
            const unsigned lo = bOr[r][wa - 1], m = bOr[r][wa], hi = bOr[r][wa + 1];
            unsigned o = m;
            o |= (m << 1) | (lo >> 31);
            o |= (m << 2) | (lo >> 30);
            o |= (m >> 1) | (hi << 31);
            o |= (m >> 2) | (hi << 30);
            rOr[r][w] = o;
        }
        {
            const unsigned lo = bAnd[r][wa - 1], m = bAnd[r][wa], hi = bAnd[r][wa + 1];
            unsigned a = m;
            a &= (m << 1) | (lo >> 31);
            a &= (m << 2) | (lo >> 30);
            a &= (m >> 1) | (hi << 31);
            a &= (m >> 2) | (hi << 30);
            rAnd[r][w] = a;
        }
    }
    __syncthreads();

    // ---- Phase 3: stream pred, fuse loss + edge weight, accumulate --------
    float sumP = 0.f, sumWt = 0.f;
    unsigned hasOne = 0u, hasZero = 0u;
    const int w   = ln >> 3;         // core word 0..3
    const int bo  = (ln & 7) * 4;    // bit offset of this lane's 4 pixels
    const int gx  = x0 + ln * 4;
    for (int k = 0; k < 8; ++k) {
        const int y = wave + k * 8;  // output row in tile, window = staged rows y..y+4
        const unsigned dil = rOr[y][w] | rOr[y+1][w] | rOr[y+2][w] | rOr[y+3][w] | rOr[y+4][w];
        const unsigned ero = rAnd[y][w] & rAnd[y+1][w] & rAnd[y+2][w] & rAnd[y+3][w] & rAnd[y+4][w];
        const unsigned edge = dil & ~ero;             // edge>0 <=> window has both classes
        const unsigned bw   = bOr[y + 2][w + 1];      // core mask word (target bits)
        const int gy = y0 + y;
        const float4 p4 = *(const float4*)(pimg + (size_t)gy * IMG_W + gx);
        const float ps[4] = {p4.x, p4.y, p4.z, p4.w};
        #pragma unroll
        for (int j = 0; j < 4; ++j) {
            const unsigned bit = (bw >> (bo + j)) & 1u;
            const float p = ps[j];
            // stable softplus(p) - p*t
            const float sp = fmaxf(p, 0.f) + log1pf(__expf(-fabsf(p)));
            const float l  = sp - p * (float)bit;
            sumP += l;
            const unsigned eb = (edge >> (bo + j)) & 1u;
            sumWt += l * (eb ? 0.1f : 1.0f);
            hasOne  |= bit;
            hasZero |= bit ^ 1u;
        }
    }

    // ---- block reduction (deterministic, no atomics) ----------------------
    #pragma unroll
    for (int off = 16; off; off >>= 1) {
        sumP  += __shfl_down(sumP,  off);
        sumWt += __shfl_down(sumWt, off);
    }
    const unsigned f = (__any((int)hasOne) ? 1u : 0u) | (__any((int)hasZero) ? 2u : 0u);
    if (ln == 0) { redP[wave] = sumP; redW[wave] = sumWt; redF[wave] = f; }
    __syncthreads();
    if (tid == 0) {
        float sp = 0.f, sw = 0.f; unsigned ff = 0u;
        #pragma unroll
        for (int i = 0; i < 8; ++i) { sp += redP[i]; sw += redW[i]; ff |= redF[i]; }
        const int b = (blockIdx.z * gridDim.y + blockIdx.y) * gridDim.x + blockIdx.x;
        wsP[b] = sp; wsW[b] = sw; wsF[b] = ff;
    }
}

// ---------------------------------------------------------------------------
// Kernel 2: one wave32. Sum 4096 partials with V_WMMA_F32_16X16X4_F32 against
// an all-ones B: D[i][j] = sum_k A[i][k] + C[i][j] -> 64-wide horizontal add
// per matrix op, accumulated in C. EXEC is all-ones (full single wave).
// ---------------------------------------------------------------------------
__global__ __launch_bounds__(32) void edge_bce_finalize_kernel(
    const float* __restrict__ wsP, const float* __restrict__ wsW,
    const unsigned* __restrict__ wsF, float* __restrict__ out)
{
    const int ln = threadIdx.x;
    // A(16x4) f32 lane layout: lanes 0-15 hold K=0,1 ; lanes 16-31 hold K=2,3; M = lane%16
    const int m  = ln & 15;
    const int kb = (ln >> 4) * 2;
    const v2f onesB = {1.0f, 1.0f};
    v8f cP = {};
    v8f cW = {};
    for (int t = 0; t < 64; ++t) {                    // 64 * 64 = 4096 partials
        const int base = t * 64 + m * 4 + kb;
        const v2f aP = {wsP[base], wsP[base + 1]};
        const v2f aW = {wsW[base], wsW[base + 1]};
        cP = __builtin_amdgcn_wmma_f32_16x16x4_f32(false, aP, false, onesB,
                                                   (short)0, cP, false, false);
        cW = __builtin_amdgcn_wmma_f32_16x16x4_f32(false, aW, false, onesB,
                                                   (short)0, cW, false, false);
    }
    // C layout: lane L, VGPR v -> C[v + 8*(L>=16)][L%16]; every column equals the
    // per-row sum, so total = sum_v ( C@lane0 + C@lane16 ).
    float tP = 0.f, tW = 0.f;
    #pragma unroll
    for (int v = 0; v < 8; ++v) {
        tP += cP[v] + __shfl(cP[v], 16);
        tW += cW[v] + __shfl(cW[v], 16);
    }
    unsigned f = 0u;
    for (int i = ln; i < NBLK; i += 32) f |= wsF[i];
    const bool cond = __any((int)(f & 1u)) && __any((int)(f & 2u));
    if (ln == 0) out[0] = (cond ? tW : tP) * (1.0f / N_TOTAL);
}

extern "C" void kernel_launch(void* const* d_in, const int* in_sizes, int n_in,
                              void* d_out, int out_size, void* d_ws, size_t ws_size,
                              hipStream_t stream) {
    (void)in_sizes; (void)n_in; (void)out_size; (void)ws_size;
    const float* pred   = (const float*)d_in[0];
    const float* target = (const float*)d_in[1];
    float* out = (float*)d_out;

    float*    wsP = (float*)d_ws;            // 4096 floats
    float*    wsW = wsP + NBLK;              // 4096 floats
    unsigned* wsF = (unsigned*)(wsW + NBLK); // 4096 u32  (48 KB total)

    dim3 grid(IMG_W / TW, IMG_H / TH, IMG_B); // 8 x 16 x 32
    edge_bce_tile_kernel<<<grid, dim3(256), 0, stream>>>(pred, target, wsP, wsW, wsF);
    edge_bce_finalize_kernel<<<dim3(1), dim3(32), 0, stream>>>(wsP, wsW, wsF, out);
}